// Graph_Perception_87084756894095
// MI455X (gfx1250) — compile-verified
//
#include <hip/hip_runtime.h>
#include <hip/hip_bf16.h>

typedef __attribute__((ext_vector_type(2))) float v2f;
typedef __attribute__((ext_vector_type(8))) float v8f;

#define D_FEAT 128
#define TILE_M 16
#define LDS_STRIDE 132   // 128 + 4 pad: conflict-free, 528B row pitch keeps 16B alignment
#define KSTEPS   (D_FEAT / 4)   // 32 WMMA K-steps per W_k

// ---------------------------------------------------------------------------
// SPMM: out[src_e] += w_e * x[dst_e]   (one wave32 per edge; lane owns 4 floats)
// x/z are L2-resident (51.2 MB each vs 192 MB L2), so gathers and atomics
// resolve in L2; HBM traffic is just the streamed edge lists.
// ---------------------------------------------------------------------------
__global__ __launch_bounds__(256) void spmm_scatter_kernel(
    const int* __restrict__ edge_src, const int* __restrict__ edge_dst,
    const float* __restrict__ edge_w, const float* __restrict__ xin,
    float* __restrict__ out, int num_e)
{
    const int e = blockIdx.x * 8 + (threadIdx.x >> 5);
    if (e >= num_e) return;
    const int lane = threadIdx.x & 31;

    const int   s  = edge_src[e];
    const int   d  = edge_dst[e];
    const float we = edge_w[e];

    // coalesced 512B row gather: 32 lanes x float4 (global_load_b128)
    const float4 v = ((const float4*)(xin + (size_t)d * D_FEAT))[lane];

    float* o = out + (size_t)s * D_FEAT + lane * 4;
    unsafeAtomicAdd(o + 0, we * v.x);   // -> global_atomic_add_f32
    unsafeAtomicAdd(o + 1, we * v.y);
    unsafeAtomicAdd(o + 2, we * v.z);
    unsafeAtomicAdd(o + 3, we * v.w);
}

// ---------------------------------------------------------------------------
// One-time repack of W (K x 128 x 128 row-major) into WMMA B-fragment order:
//   Wp[(k*KSTEPS + kk)*2 + hi][n] = { W[k][4kk+2hi][n], W[k][4kk+2hi+1][n] }
// so each lane's (b.x, b.y) pair is one contiguous, coalesced global_load_b64.
// Total: 3*32*2*128 = 24576 float2 (196 KB), L2-resident across all blocks.
// ---------------------------------------------------------------------------
__global__ __launch_bounds__(256) void repack_W_kernel(
    const float* __restrict__ W, v2f* __restrict__ Wp)
{
    const int i = blockIdx.x * 256 + threadIdx.x;   // [0, 24576)
    if (i >= 3 * KSTEPS * 2 * D_FEAT) return;
    const int n  = i & (D_FEAT - 1);
    const int t  = i >> 7;          // k*64 + kk*2 + hi
    const int hi = t & 1;
    const int kk = (t >> 1) & (KSTEPS - 1);
    const int k  = t >> 6;
    const int kd = kk * 4 + hi * 2;
    const float* Wk = W + (size_t)k * D_FEAT * D_FEAT;
    v2f p;
    p.x = Wk[(size_t)kd * D_FEAT + n];
    p.y = Wk[(size_t)(kd + 1) * D_FEAT + n];
    Wp[i] = p;
}

// ---------------------------------------------------------------------------
// Dense reduction + PReLU:
//   y[n][f] = prelu( sum_k sum_d Z_k[n][d] * W[k][d][f] )
// Block = 256 threads (8 waves), tile = 16 nodes x 128 output features.
// Each wave computes a 16x16 tile via 96x V_WMMA_F32_16X16X4_F32.
// ---------------------------------------------------------------------------
__global__ __launch_bounds__(256) void dense_prelu_kernel(
    const float* __restrict__ x,  const float* __restrict__ z1,
    const float* __restrict__ z2, const v2f* __restrict__ Wp,
    const float* __restrict__ prelu_a, float* __restrict__ out, int num_n)
{
    __shared__ __align__(16) float zt[TILE_M * LDS_STRIDE];

    const int tid      = threadIdx.x;
    const int wave     = tid >> 5;
    const int lane     = tid & 31;
    const int hi       = lane >> 4;   // lane half: K pair select (A/B); row+8 (C/D)
    const int mn       = lane & 15;   // A row / B col / D col
    const long node_base = (long)blockIdx.x * TILE_M;
    const int  col_base  = wave * 16;
    const bool full_tile = (node_base + TILE_M) <= (long)num_n;

    v8f acc = {};   // 16x16 f32 accumulator (8 VGPRs)

    const float* Zs[3] = { x, z1, z2 };

    #pragma unroll
    for (int k = 0; k < 3; ++k) {
        __syncthreads();
        if (full_tile) {
            // fast path: 512 float4s, coalesced global_load_b128 -> ds_store_b128
            const float4* zsrc4 = (const float4*)(Zs[k] + node_base * D_FEAT);
            #pragma unroll
            for (int i = tid; i < TILE_M * D_FEAT / 4; i += 256) {
                const int r = i >> 5;          // / (128/4)
                const int c = i & 31;
                *(float4*)&zt[r * LDS_STRIDE + c * 4] = zsrc4[i];
            }
        } else {
            // tail path: element-wise with zero padding
            const float* zsrc = Zs[k] + node_base * D_FEAT;
            for (int i = tid; i < TILE_M * D_FEAT; i += 256) {
                const int r = i >> 7;
                const int c = i & (D_FEAT - 1);
                zt[r * LDS_STRIDE + c] = (node_base + r < num_n) ? zsrc[i] : 0.0f;
            }
        }
        __syncthreads();

        // fragment-ordered B pointer: ((k*32 + kk)*2 + hi)*128 + col
        const v2f* bptr = Wp + ((size_t)(k * 2 * KSTEPS + hi) * D_FEAT)
                             + col_base + mn;
        const float* arow = &zt[mn * LDS_STRIDE + hi * 2];
        #pragma unroll
        for (int kk = 0; kk < KSTEPS; ++kk) {
            v2f a = *(const v2f*)(arow + kk * 4);       // ds_load_b64
            v2f b = bptr[(size_t)kk * 2 * D_FEAT];      // global_load_b64, coalesced
            acc = __builtin_amdgcn_wmma_f32_16x16x4_f32(
                      /*neg_a=*/false, a, /*neg_b=*/false, b,
                      /*c_mod=*/(short)0, acc,
                      /*reuse_a=*/false, /*reuse_b=*/false);
        }
    }

    // PReLU + store. D layout: VGPR r -> row (r + 8*hi), col mn.
    const float slope = prelu_a[0];
    const long row0 = node_base + hi * 8;
    float* obase = out + row0 * D_FEAT + col_base + mn;
    if (full_tile) {
        #pragma unroll
        for (int r = 0; r < 8; ++r) {
            float v = acc[r];
            v = (v >= 0.0f) ? v : slope * v;
            obase[(size_t)r * D_FEAT] = v;
        }
    } else {
        #pragma unroll
        for (int r = 0; r < 8; ++r) {
            if (row0 + r < num_n) {
                float v = acc[r];
                v = (v >= 0.0f) ? v : slope * v;
                obase[(size_t)r * D_FEAT] = v;
            }
        }
    }
}

// ---------------------------------------------------------------------------
// Launcher
// Inputs (setup_inputs order): x[N*D] f32, W[K*D*D] f32, prelu_a[1] f32,
//                              edge_weight[E] f32, edge_src[E] i32, edge_dst[E] i32
// Output: y[N*D] f32
// Workspace: z1[N*D] | z2[N*D] | Wp[24576 float2]
// ---------------------------------------------------------------------------
extern "C" void kernel_launch(void* const* d_in, const int* in_sizes, int n_in,
                              void* d_out, int out_size, void* d_ws, size_t ws_size,
                              hipStream_t stream)
{
    const float* x       = (const float*)d_in[0];
    const float* W       = (const float*)d_in[1];
    const float* prelu_a = (const float*)d_in[2];
    const float* ew      = (const float*)d_in[3];
    const int*   es      = (const int*)d_in[4];
    const int*   ed      = (const int*)d_in[5];
    float*       out     = (float*)d_out;

    const int num_n = in_sizes[0] / D_FEAT;   // N = 100000
    const int num_e = in_sizes[3];            // E = 3200000

    float* z1 = (float*)d_ws;
    float* z2 = z1 + (size_t)num_n * D_FEAT;
    v2f*   Wp = (v2f*)(z2 + (size_t)num_n * D_FEAT);

    // z buffers must start at zero for the atomic scatter-add
    hipMemsetAsync(d_ws, 0, 2 * (size_t)num_n * D_FEAT * sizeof(float), stream);

    // one-time W repack into WMMA B-fragment order (96 blocks, negligible)
    repack_W_kernel<<<(3 * KSTEPS * 2 * D_FEAT + 255) / 256, 256, 0, stream>>>(W, Wp);

    const int eblocks = (num_e + 7) / 8;      // 8 edges per 256-thread block
    spmm_scatter_kernel<<<eblocks, 256, 0, stream>>>(es, ed, ew, x,  z1, num_e);
    spmm_scatter_kernel<<<eblocks, 256, 0, stream>>>(es, ed, ew, z1, z2, num_e);

    const int nblocks = (num_n + TILE_M - 1) / TILE_M;
    dense_prelu_kernel<<<nblocks, 256, 0, stream>>>(x, z1, z2, Wp, prelu_a, out, num_n);
}